// BSplineActivation_63660005261530
// MI455X (gfx1250) — compile-verified
//
#include <hip/hip_runtime.h>

// B-spline (KAN-style) activation for MI455X (gfx1250).
// Memory-bound streaming kernel: 268 MB single-pass traffic -> ~11.5 us floor
// at 23.3 TB/s. Einsum folded into a 3-entry piecewise-linear table evaluated
// with two v_med3_num_f32 + two FMAs; SiLU via hardware v_tanh_f32.
// B128 non-temporal loads/stores + global_prefetch_b8.

typedef float v4f __attribute__((ext_vector_type(4)));

__device__ __forceinline__ float fmed3(float a, float b, float c) {
#if __has_builtin(__builtin_amdgcn_fmed3f)
    return __builtin_amdgcn_fmed3f(a, b, c);
#else
    return fminf(fmaxf(a, b), c);   // b<=c assumed by callers
#endif
}

__device__ __forceinline__ float bspline_elem(float xi, float s1,
                                              float d01, float d12, float hbw) {
    // spline_weight * y_spline = s1 + med3(x,-1,0)*d01 + med3(x,0,1)*d12
    float m1 = fmed3(xi, -1.0f, 0.0f);
    float m2 = fmed3(xi,  0.0f, 1.0f);
    float ys = fmaf(m2, d12, fmaf(m1, d01, s1));

    // base_weight * x * sigmoid(x) = hb*(1 + tanh(x/2)), hb = 0.5*bw*x
    float hb = hbw * xi;
#if __has_builtin(__builtin_amdgcn_tanhf)
    float th = __builtin_amdgcn_tanhf(0.5f * xi);
#else
    float e  = __expf(-xi);                                   // v_exp_f32
    float th = (1.0f - e) * __builtin_amdgcn_rcpf(1.0f + e);  // tanh(x/2)
#endif
    return fmaf(hb, th, hb + ys);
}

__global__ __launch_bounds__(256) void bspline_main(
    const float* __restrict__ x, const float* __restrict__ cp,
    const float* __restrict__ bwp, const float* __restrict__ swp,
    const float* __restrict__ basis, float* __restrict__ out, long long n8)
{
    // Fold control points into per-grid-point scalars (uniform -> SGPR math).
    float c0 = cp[0], c1 = cp[1], c2 = cp[2], c3 = cp[3], c4 = cp[4];
    float sw = swp[0];
    float t0 = fmaf(basis[4],  c4, fmaf(basis[3],  c3, fmaf(basis[2],  c2, fmaf(basis[1],  c1, basis[0]  * c0))));
    float t1 = fmaf(basis[9],  c4, fmaf(basis[8],  c3, fmaf(basis[7],  c2, fmaf(basis[6],  c1, basis[5]  * c0))));
    float t2 = fmaf(basis[14], c4, fmaf(basis[13], c3, fmaf(basis[12], c2, fmaf(basis[11], c1, basis[10] * c0))));
    float s0 = sw * t0, s1 = sw * t1, s2 = sw * t2;
    float d01 = s1 - s0, d12 = s2 - s1;
    float hbw = 0.5f * bwp[0];

    long long i      = (long long)blockIdx.x * blockDim.x + threadIdx.x;
    long long stride = (long long)gridDim.x * blockDim.x;
    for (; i < n8; i += stride) {
        const v4f* src = (const v4f*)x + 2 * i;
        // Two independent B128 non-temporal loads per iteration (MLP depth).
        v4f xa = __builtin_nontemporal_load(src);
        v4f xb = __builtin_nontemporal_load(src + 1);
        // Pull upcoming lines toward L2 (global_prefetch_b8).
        __builtin_prefetch((const char*)src + 32768, 0, 0);

        v4f oa, ob;
#pragma unroll
        for (int k = 0; k < 4; ++k) {
            oa[k] = bspline_elem(xa[k], s1, d01, d12, hbw);
            ob[k] = bspline_elem(xb[k], s1, d01, d12, hbw);
        }
        v4f* dst = (v4f*)out + 2 * i;
        __builtin_nontemporal_store(oa, dst);
        __builtin_nontemporal_store(ob, dst + 1);
    }
}

// Scalar tail for n % 8 != 0 (not hit for 8*2048*2048, kept for generality).
__global__ void bspline_tail(
    const float* __restrict__ x, const float* __restrict__ cp,
    const float* __restrict__ bwp, const float* __restrict__ swp,
    const float* __restrict__ basis, float* __restrict__ out,
    long long start, long long n)
{
    float c0 = cp[0], c1 = cp[1], c2 = cp[2], c3 = cp[3], c4 = cp[4];
    float sw = swp[0];
    float t0 = fmaf(basis[4],  c4, fmaf(basis[3],  c3, fmaf(basis[2],  c2, fmaf(basis[1],  c1, basis[0]  * c0))));
    float t1 = fmaf(basis[9],  c4, fmaf(basis[8],  c3, fmaf(basis[7],  c2, fmaf(basis[6],  c1, basis[5]  * c0))));
    float t2 = fmaf(basis[14], c4, fmaf(basis[13], c3, fmaf(basis[12], c2, fmaf(basis[11], c1, basis[10] * c0))));
    float s0 = sw * t0, s1 = sw * t1, s2 = sw * t2;
    float d01 = s1 - s0, d12 = s2 - s1;
    float hbw = 0.5f * bwp[0];

    long long i = start + (long long)blockIdx.x * blockDim.x + threadIdx.x;
    if (i < n) out[i] = bspline_elem(x[i], s1, d01, d12, hbw);
}

extern "C" void kernel_launch(void* const* d_in, const int* in_sizes, int n_in,
                              void* d_out, int out_size, void* d_ws, size_t ws_size,
                              hipStream_t stream) {
    const float* x     = (const float*)d_in[0];
    const float* cp    = (const float*)d_in[1];
    const float* bw    = (const float*)d_in[2];
    const float* sw    = (const float*)d_in[3];
    const float* basis = (const float*)d_in[4];
    float* out = (float*)d_out;

    long long n  = (long long)in_sizes[0];
    long long n8 = n / 8;               // 8 floats (2 x B128) per thread-iter
    if (n8 > 0) {
        const int block = 256;          // 8 wave32s per block
        long long want   = (n8 + 1) / 2;  // ~2 iterations per thread
        long long blocks = (want + block - 1) / block;
        if (blocks < 1) blocks = 1;
        bspline_main<<<dim3((unsigned)blocks), dim3(block), 0, stream>>>(
            x, cp, bw, sw, basis, out, n8);
    }
    long long done = n8 * 8;
    long long rem  = n - done;
    if (rem > 0) {
        const int block = 256;
        long long blocks = (rem + block - 1) / block;
        bspline_tail<<<dim3((unsigned)blocks), dim3(block), 0, stream>>>(
            x, cp, bw, sw, basis, out, done, n);
    }
}